// SAM3DPointCloudBuilder_61718680043522
// MI455X (gfx1250) — compile-verified
//
#include <hip/hip_runtime.h>

// SAM3D point-cloud back-projection: out[b,y,x,:] = valid ? ((x-cx)z/fx, (y-cy)z/fy, z) : 0
// Pure HBM-streaming kernel (AI ~0.4 FLOP/B, ~14.5us floor @ 23.3 TB/s): optimized for
// MI455X with 128-bit vector memory, non-temporal cache hints, global_prefetch_b8,
// wave32 block sizing. WMMA/TDM intentionally unused: zero operand reuse, bandwidth roof.

#define IMG_H 512
#define IMG_W 512
#define PIX_PER_THREAD 4
#define BLOCK_THREADS 256   // 8 wave32 per block

typedef float v4f __attribute__((ext_vector_type(4)));  // clang-native vector for NT builtins

__global__ __launch_bounds__(BLOCK_THREADS) void
sam3d_backproject_kernel(const float* __restrict__ depth,
                         const float* __restrict__ masks,
                         const float* __restrict__ intr,
                         float* __restrict__ out)
{
    const int HW  = IMG_H * IMG_W;
    const int b   = blockIdx.y;                                  // batch (uniform)
    const int tid = blockIdx.x * BLOCK_THREADS + threadIdx.x;
    const int p0  = tid * PIX_PER_THREAD;                        // first pixel in image
    if (p0 >= HW) return;

    // Uniform per-batch intrinsics -> scalar loads (s_load_*).
    const float* __restrict__ Kb = intr + (size_t)b * 9;
    const float fx = Kb[0], cx = Kb[2], fy = Kb[4], cy = Kb[5];
    const float rfx = 1.0f / fx;
    const float rfy = 1.0f / fy;

    const int y  = p0 >> 9;            // p0 / 512 ; 4 consecutive pixels never cross a row
    const int x0 = p0 & (IMG_W - 1);

    const size_t base = (size_t)b * HW + (size_t)p0;

    // 128-bit non-temporal streaming loads (once-touched data; don't pollute L2/WGP$).
    v4f d4 = __builtin_nontemporal_load((const v4f*)(depth + base));
    v4f m4 = __builtin_nontemporal_load((const v4f*)(masks + base));

    // Speculative distance prefetch into GL2 (gfx1250 global_prefetch_b8).
    // 16 KiB ahead within this batch's stream; OOB translation failures are dropped.
    __builtin_prefetch(depth + base + 4096, 0, 1);
    __builtin_prefetch(masks + base + 4096, 0, 1);

    const float dd[4] = {d4.x, d4.y, d4.z, d4.w};
    const float mm[4] = {m4.x, m4.y, m4.z, m4.w};
    const float py_row = ((float)y - cy) * rfy;   // shared by the 4 pixels (same row)

    float o[12];
#pragma unroll
    for (int i = 0; i < PIX_PER_THREAD; ++i) {
        const float d = dd[i];
        const bool valid = (mm[i] > 0.5f) && __builtin_isfinite(d) && (d > 0.0f);
        const float px = ((float)(x0 + i) - cx) * rfx * d;
        const float py = py_row * d;
        // Select AFTER arithmetic: invalid (incl. NaN/Inf depth) -> exact 0, like jnp.where.
        o[3 * i + 0] = valid ? px : 0.0f;
        o[3 * i + 1] = valid ? py : 0.0f;
        o[3 * i + 2] = valid ? d  : 0.0f;
    }

    // 12 contiguous floats -> 3x 128-bit non-temporal stores (48 B per thread;
    // a wave covers a fully dense 1536 B span, so L2 sees only full-line writes).
    float* __restrict__ op = out + base * 3;
    v4f s0 = {o[0], o[1],  o[2],  o[3]};
    v4f s1 = {o[4], o[5],  o[6],  o[7]};
    v4f s2 = {o[8], o[9],  o[10], o[11]};
    __builtin_nontemporal_store(s0, (v4f*)(op + 0));
    __builtin_nontemporal_store(s1, (v4f*)(op + 4));
    __builtin_nontemporal_store(s2, (v4f*)(op + 8));
}

extern "C" void kernel_launch(void* const* d_in, const int* in_sizes, int n_in,
                              void* d_out, int out_size, void* d_ws, size_t ws_size,
                              hipStream_t stream)
{
    (void)n_in; (void)out_size; (void)d_ws; (void)ws_size;

    const float* depth = (const float*)d_in[0];
    const float* masks = (const float*)d_in[1];
    const float* intr  = (const float*)d_in[2];
    float*       out   = (float*)d_out;

    const int HW = IMG_H * IMG_W;
    const int B  = in_sizes[0] / HW;                               // 64
    const int blocks_x = HW / (PIX_PER_THREAD * BLOCK_THREADS);    // 256

    dim3 grid(blocks_x, B, 1);
    dim3 block(BLOCK_THREADS, 1, 1);
    sam3d_backproject_kernel<<<grid, block, 0, stream>>>(depth, masks, intr, out);
}